// MultiHeadedAttetion_781684048500
// MI455X (gfx1250) — compile-verified
//
#include <hip/hip_runtime.h>
#include <hip/hip_bf16.h>

// ---------------------------------------------------------------------------
// MultiHeadedAttention forward for MI455X (gfx1250), f32 WMMA path.
// B=2, S=2048, E=128 (head dim), H=16, D=2048.
// Output = [ out (B*S*D f32) | weights (B*H*S*S f32) ] concatenated.
//
// Round 3: fix async-to-LDS builtin signature (v4i AS1/AS3 pointers).
// ---------------------------------------------------------------------------

typedef __attribute__((ext_vector_type(2))) float v2f;
typedef __attribute__((ext_vector_type(8))) float v8f;
typedef __attribute__((ext_vector_type(4))) unsigned int u32x4;
typedef __attribute__((ext_vector_type(8))) unsigned int u32x8;
typedef int v4i __attribute__((vector_size(4 * sizeof(int))));

constexpr int B_ = 2;
constexpr int S_ = 2048;
constexpr int E_ = 128;
constexpr int H_ = 16;
constexpr int D_ = 2048;
constexpr float SCALE = 0.08838834764831845f;  // 1/sqrt(128)

#ifndef __has_builtin
#define __has_builtin(x) 0
#endif

#if __has_builtin(__builtin_amdgcn_global_load_async_to_lds_b128)
#define HAVE_ASYNC_LDS 1
#else
#define HAVE_ASYNC_LDS 0
#endif

// AS1 = global ("__device__"), AS3 = LDS ("__shared__") pointers to 16B vecs
#define AS1_V4(p) ((__attribute__((address_space(1))) v4i*)(p))
#define AS3_V4(p) ((__attribute__((address_space(3))) v4i*)(p))

static __device__ __forceinline__ void wait_asynccnt0() {
#if __has_builtin(__builtin_amdgcn_s_wait_asynccnt)
  __builtin_amdgcn_s_wait_asynccnt(0);
#else
  asm volatile("s_wait_asynccnt 0" ::: "memory");
#endif
}

static __device__ __forceinline__ void wait_tensorcnt0() {
#if __has_builtin(__builtin_amdgcn_s_wait_tensorcnt)
  __builtin_amdgcn_s_wait_tensorcnt(0);
#else
  asm volatile("s_wait_tensorcnt 0" ::: "memory");
#endif
}

static __device__ __forceinline__ v8f wmma_f32(v2f a, v2f b, v8f c) {
  // D = A(16x4,f32) * B(4x16,f32) + C(16x16,f32)
  return __builtin_amdgcn_wmma_f32_16x16x4_f32(false, a, false, b,
                                               (short)0, c, false, false);
}

static __device__ __forceinline__ v8f zero8() {
  v8f z = {0.f, 0.f, 0.f, 0.f, 0.f, 0.f, 0.f, 0.f};
  return z;
}

// Issue one TDM 2D tile load: 128 rows x 16 f32 cols, row stride D_ floats,
// from global 'gsrc' into LDS at 'lds_dst'.  Wave-uniform; caller gates to
// one wave and syncs with s_wait_tensorcnt + workgroup barrier.
static __device__ __forceinline__ void tdm_load_wo_tile(const float* gsrc,
                                                        float* lds_dst) {
  unsigned lds_off =
      (unsigned)(size_t)(__attribute__((address_space(3))) char*)lds_dst;
  unsigned long long ga = (unsigned long long)(size_t)gsrc;

  u32x4 g0;
  g0.x = 1u;                                            // count=1, flags=0
  g0.y = lds_off;                                       // lds_addr
  g0.z = (unsigned)ga;                                  // global_addr[31:0]
  g0.w = (unsigned)((ga >> 32) & 0x1FFFFFFu) | (2u << 30);  // [56:32]|type=2

  u32x8 g1;
  g1.s0 = (2u << 16);        // workgroup_mask=0, data_size=2 (4 bytes)
  g1.s1 = (16u << 16);       // tensor_dim0 = 16 (low 16 bits at 63:48)
  g1.s2 = (128u << 16);      // tensor_dim0_hi=0 | tensor_dim1 = 128
  g1.s3 = (16u << 16);       // tensor_dim1_hi=0 | tile_dim0 = 16
  g1.s4 = 128u;              // tile_dim1 = 128, tile_dim2 = 0
  g1.s5 = (unsigned)D_;      // tensor_dim0_stride = 2048 (low 32)
  g1.s6 = 0u;                // stride0_hi=0, stride1_lo=0
  g1.s7 = 0u;

  asm volatile("tensor_load_to_lds %0, %1" ::"s"(g0), "s"(g1) : "memory");
}

// ---------------------------------------------------------------------------
// Kernel 1: X[B*S,E] @ W[E,D] -> head-major [B,H,S,E].  grid.z selects q/k/v.
// Block = 4 waves sharing one 16-col W tile staged in LDS (async copy);
// each wave computes one 16x16 tile via 32 f32 WMMAs.
// ---------------------------------------------------------------------------
__global__ void __launch_bounds__(128) proj_kernel(
    const float* __restrict__ Xq, const float* __restrict__ Xk,
    const float* __restrict__ Xv, const float* __restrict__ Wq,
    const float* __restrict__ Wk, const float* __restrict__ Wv,
    float* __restrict__ qh, float* __restrict__ kh, float* __restrict__ vh) {
  __shared__ float wlds[128][16];  // 8 KB: K=128 x N=16 slab of W

  const float* X;
  const float* W;
  float* Out;
  if (blockIdx.z == 0) { X = Xq; W = Wq; Out = qh; }
  else if (blockIdx.z == 1) { X = Xk; W = Wk; Out = kh; }
  else { X = Xv; W = Wv; Out = vh; }

  const int tilesN = D_ / 16;  // 128
  const int mb = blockIdx.x / tilesN;
  const int tn = blockIdx.x % tilesN;
  const int w = threadIdx.x >> 5;
  const int lane = threadIdx.x & 31;
  const int hf = lane >> 4;
  const int lr = lane & 15;
  const int m0 = (mb * 4 + w) * 16;
  const int n0 = tn * 16;

  // cooperative stage of W[:, n0:n0+16): thread t copies row t (64 bytes)
  {
    const float* src = W + (size_t)threadIdx.x * D_ + n0;
    float* dst = &wlds[threadIdx.x][0];
#if HAVE_ASYNC_LDS
#pragma unroll
    for (int c = 0; c < 4; ++c) {
      __builtin_amdgcn_global_load_async_to_lds_b128(AS1_V4(src + c * 4),
                                                     AS3_V4(dst + c * 4), 0, 0);
    }
    wait_asynccnt0();
#else
#pragma unroll
    for (int c = 0; c < 16; ++c) dst[c] = src[c];
#endif
  }
  __syncthreads();

  v8f acc = zero8();
  const float* xrow = X + (size_t)(m0 + lr) * E_;
#pragma unroll 8
  for (int kk = 0; kk < E_; kk += 4) {
    const int ka = kk + hf * 2;
    v2f a, b;
    a.x = xrow[ka];
    a.y = xrow[ka + 1];
    b.x = wlds[ka][lr];
    b.y = wlds[ka + 1][lr];
    acc = wmma_f32(a, b, acc);
  }

  // scatter to head-major [B,H,S,E]
  const int n = n0 + lr, h = n >> 7, e = n & 127;
#pragma unroll
  for (int r = 0; r < 8; ++r) {
    const int m = m0 + r + hf * 8;
    const int bb = m >> 11, s = m & (S_ - 1);
    Out[(((size_t)(bb * H_ + h) * S_ + s) * E_) + e] = acc[r];
  }
}

// ---------------------------------------------------------------------------
// Kernel 2: attention core. weights[i,j] = softmax_j(k_i . q_j / sqrt(E)),
// masked j<i (exact 0), then O = weights @ V.  One wave owns 16 rows (i) of
// one (b,h); two passes (running max/sum, then emit P + P@V).
// ---------------------------------------------------------------------------
__global__ void __launch_bounds__(128) attn_kernel(
    const float* __restrict__ qh, const float* __restrict__ kh,
    const float* __restrict__ vh, float* __restrict__ wout,
    float* __restrict__ att) {
  __shared__ float pbuf[4][16][17];  // per-wave P tile, padded vs bank conflicts

  const int w = threadIdx.x >> 5;
  const int lane = threadIdx.x & 31;
  const int hf = lane >> 4;
  const int lr = lane & 15;

  const int iblk = blockIdx.x & 31;  // S/64 = 32 row-blocks
  const int bh = blockIdx.x >> 5;    // b*H + h
  const int i0 = iblk * 64 + w * 16;

  const float* Kp = kh + (size_t)bh * S_ * E_;
  const float* Qp = qh + (size_t)bh * S_ * E_;
  const float* Vp = vh + (size_t)bh * S_ * E_;
  float* Wp = wout + (size_t)bh * S_ * S_;

  // Resident A-fragments of this wave's 16 K rows (16x128 f32 -> 64 VGPRs).
  v2f kf[32];
  const float* krow = Kp + (size_t)(i0 + lr) * E_;
#pragma unroll
  for (int c = 0; c < 32; ++c) {
    const int ka = c * 4 + hf * 2;
    kf[c].x = krow[ka];
    kf[c].y = krow[ka + 1];
  }

  float rm[8], rs[8];
#pragma unroll
  for (int r = 0; r < 8; ++r) { rm[r] = -3.0e38f; rs[r] = 0.0f; }

  const int jt0 = i0 >> 4;

  // ---- pass 1: running row max / row sumexp over unmasked j tiles ----
  for (int jt = jt0; jt < S_ / 16; ++jt) {
    const int j0 = jt * 16;
    v8f c = zero8();
    const float* qrow = Qp + (size_t)(j0 + lr) * E_;
#pragma unroll
    for (int kk = 0; kk < 32; ++kk) {
      const int ka = kk * 4 + hf * 2;
      v2f bf;
      bf.x = qrow[ka];
      bf.y = qrow[ka + 1];
      c = wmma_f32(kf[kk], bf, c);
    }
#pragma unroll
    for (int r = 0; r < 8; ++r) {
      const int i = i0 + r + hf * 8;
      const int j = j0 + lr;
      float sc = c[r] * SCALE;
      if (j < i) sc = -1.0e7f;
      float tmax = sc;
      tmax = fmaxf(tmax, __shfl_xor(tmax, 1));
      tmax = fmaxf(tmax, __shfl_xor(tmax, 2));
      tmax = fmaxf(tmax, __shfl_xor(tmax, 4));
      tmax = fmaxf(tmax, __shfl_xor(tmax, 8));
      const float nm = fmaxf(rm[r], tmax);
      float ts = __expf(sc - nm);  // masked -> exp(-1e7-nm) == 0 exactly
      ts += __shfl_xor(ts, 1);
      ts += __shfl_xor(ts, 2);
      ts += __shfl_xor(ts, 4);
      ts += __shfl_xor(ts, 8);
      rs[r] = rs[r] * __expf(rm[r] - nm) + ts;
      rm[r] = nm;
    }
  }
  float rr[8];
#pragma unroll
  for (int r = 0; r < 8; ++r) rr[r] = 1.0f / rs[r];

  // ---- masked (strictly lower) tiles: exact zeros, streamed NT ----
  for (int jt = 0; jt < jt0; ++jt) {
    const int j = jt * 16 + lr;
#pragma unroll
    for (int r = 0; r < 8; ++r) {
      const int i = i0 + r + hf * 8;
      __builtin_nontemporal_store(0.0f, &Wp[(size_t)i * S_ + j]);
    }
  }

  // ---- pass 2: emit normalized P (NT store) and accumulate O += P @ V ----
  v8f o[8];
#pragma unroll
  for (int nt = 0; nt < 8; ++nt) o[nt] = zero8();

  for (int jt = jt0; jt < S_ / 16; ++jt) {
    const int j0 = jt * 16;
    v8f c = zero8();
    const float* qrow = Qp + (size_t)(j0 + lr) * E_;
#pragma unroll
    for (int kk = 0; kk < 32; ++kk) {
      const int ka = kk * 4 + hf * 2;
      v2f bf;
      bf.x = qrow[ka];
      bf.y = qrow[ka + 1];
      c = wmma_f32(kf[kk], bf, c);
    }
#pragma unroll
    for (int r = 0; r < 8; ++r) {
      const int i = i0 + r + hf * 8;
      const int j = j0 + lr;
      float sc = c[r] * SCALE;
      if (j < i) sc = -1.0e7f;
      const float p = __expf(sc - rm[r]) * rr[r];
      __builtin_nontemporal_store(p, &Wp[(size_t)i * S_ + j]);
      pbuf[w][r + hf * 8][lr] = p;  // stage for layout change (C -> A frag)
    }
    // same-wave LDS RAW: wait DS counter, block compiler reordering
    asm volatile("s_wait_dscnt 0" ::: "memory");
#pragma unroll
    for (int cc = 0; cc < 4; ++cc) {
      const int kb = cc * 4 + hf * 2;
      v2f a;
      a.x = pbuf[w][lr][kb];
      a.y = pbuf[w][lr][kb + 1];
      const float* vr0 = Vp + (size_t)(j0 + kb) * E_;
      const float* vr1 = Vp + (size_t)(j0 + kb + 1) * E_;
#pragma unroll
      for (int nt = 0; nt < 8; ++nt) {
        v2f bv;
        bv.x = vr0[nt * 16 + lr];
        bv.y = vr1[nt * 16 + lr];
        o[nt] = wmma_f32(a, bv, o[nt]);
      }
    }
  }

  // ---- store O (16 x 128) into att[B,S,D], D index = h*128 + e ----
  const int b0 = bh >> 4, hh = bh & 15;
#pragma unroll
  for (int nt = 0; nt < 8; ++nt) {
    const int e = nt * 16 + lr;
#pragma unroll
    for (int r = 0; r < 8; ++r) {
      const int i = i0 + r + hf * 8;
      att[(size_t)(b0 * S_ + i) * D_ + hh * E_ + e] = o[nt][r];
    }
  }
}

// ---------------------------------------------------------------------------
// Kernel 3: att[B*S,D] @ Wo[D,D] -> out[B*S,D].  Block = 4 waves sharing one
// 16-col Wo slab; K=2048 in 16 chunks of 128, TDM double-buffered into LDS.
// ---------------------------------------------------------------------------
__global__ void __launch_bounds__(128) outproj_kernel(
    const float* __restrict__ att, const float* __restrict__ Wo,
    float* __restrict__ out) {
  __shared__ float wlds[2][128][16];  // 16 KB double buffer

  const int tilesN = D_ / 16;  // 128
  const int mb = blockIdx.x / tilesN;
  const int tn = blockIdx.x % tilesN;
  const int w = threadIdx.x >> 5;
  const int lane = threadIdx.x & 31;
  const int hf = lane >> 4;
  const int lr = lane & 15;
  const int m0 = (mb * 4 + w) * 16;
  const int n0 = tn * 16;

  const float* arow = att + (size_t)(m0 + lr) * D_;
  v8f acc = zero8();

  // prologue: wave 0 kicks off chunk 0 via the Tensor Data Mover
  if (w == 0) tdm_load_wo_tile(Wo + (size_t)n0, &wlds[0][0][0]);

  for (int ck = 0; ck < 16; ++ck) {
    const int cur = ck & 1;
    if (w == 0) wait_tensorcnt0();  // chunk ck landed in wlds[cur]
    __syncthreads();                // publish to all waves
    if (w == 0 && ck + 1 < 16) {
      tdm_load_wo_tile(Wo + (size_t)(ck + 1) * 128 * D_ + n0,
                       &wlds[1 - cur][0][0]);
    }
    if (ck + 1 < 16) {
      __builtin_prefetch(arow + (size_t)(ck + 1) * 128, 0, 1);
    }
#pragma unroll 8
    for (int kk = 0; kk < 128; kk += 4) {
      const int ka = kk + hf * 2;
      v2f a, b;
      a.x = arow[ck * 128 + ka];
      a.y = arow[ck * 128 + ka + 1];
      b.x = wlds[cur][ka][lr];
      b.y = wlds[cur][ka + 1][lr];
      acc = wmma_f32(a, b, acc);
    }
    __syncthreads();  // all waves done with wlds[cur] before it is reloaded
  }

#pragma unroll
  for (int r = 0; r < 8; ++r) {
    const int m = m0 + r + hf * 8;
    out[(size_t)m * D_ + n0 + lr] = acc[r];
  }
}

// ---------------------------------------------------------------------------
extern "C" void kernel_launch(void* const* d_in, const int* in_sizes, int n_in,
                              void* d_out, int out_size, void* d_ws,
                              size_t ws_size, hipStream_t stream) {
  (void)in_sizes; (void)n_in; (void)out_size; (void)ws_size;
  const float* Q  = (const float*)d_in[0];
  const float* K  = (const float*)d_in[1];
  const float* V  = (const float*)d_in[2];
  const float* Wq = (const float*)d_in[3];
  const float* Wk = (const float*)d_in[4];
  const float* Wv = (const float*)d_in[5];
  const float* Wo = (const float*)d_in[6];

  float* out = (float*)d_out;
  float* weights = out + (size_t)B_ * S_ * D_;  // second tuple element

  const size_t per = (size_t)B_ * H_ * S_ * E_;  // 8,388,608 floats
  float* ws = (float*)d_ws;
  float* qh = ws;
  float* kh = ws + per;
  float* vh = ws + 2 * per;
  float* att = ws + 3 * per;  // B*S*D floats

  // 1) q/k/v projections into head-major layout (async W staging)
  {
    dim3 grid((B_ * S_ / 64) * (D_ / 16), 1, 3);  // 8192 x 1 x 3
    proj_kernel<<<grid, 128, 0, stream>>>(Q, K, V, Wq, Wk, Wv, qh, kh, vh);
  }
  // 2) attention core: weights (NT-streamed, 512 MB) + O = P@V
  attn_kernel<<<dim3(B_ * H_ * (S_ / 64)), 128, 0, stream>>>(qh, kh, vh,
                                                             weights, att);
  // 3) output projection (TDM double-buffered Wo staging)
  outproj_kernel<<<dim3((B_ * S_ / 64) * (D_ / 16)), 128, 0, stream>>>(
      att, Wo, out);
}